// TokenMergeModule_76845554860101
// MI455X (gfx1250) — compile-verified
//
#include <hip/hip_runtime.h>

// Problem constants (baked from setup_inputs)
#define B_    4
#define S_    4096
#define D_    768
#define G_    64
#define W_    16
#define R_    4
#define NW_   (S_ / W_)        // 256 windows per batch row
#define NWIN_ (B_ * NW_)       // 1024 windows total
#define WOUT_ (W_ - R_)        // 12 tokens kept per window
#define NOUT_ (NWIN_ * WOUT_)  // 12288 output tokens

#define EPS_NORM  1e-12f
#define EPS_MERGE 1e-8f

typedef __attribute__((ext_vector_type(2))) float v2f;
typedef __attribute__((ext_vector_type(8))) float v8f;

struct MergeInfo {
    int   src0;   // global token row of kept token
    int   src1;   // global token row of merged partner, or -1
    float w0;     // weight for src0 (1.0 if unmerged)
    float w1;     // weight for src1 (0.0 if unmerged)
};

// ---------------------------------------------------------------------------
// Kernel 1: per-window gating GEMM (WMMA f32 16x16x4), norms, adjacent cosine
// sims, greedy pair selection, and emission of merge descriptors + positions.
// One block (4 waves, 128 threads) per 16-token window. Each wave owns one
// 16x16 N-tile of g (G=64 -> 4 tiles), chained over K=768 in steps of 4.
// ---------------------------------------------------------------------------
__global__ __launch_bounds__(128) void tm_select_kernel(
    const float* __restrict__ x,    // (B*S, D)
    const float* __restrict__ Wg,   // (D, G)
    const int*   __restrict__ pos,  // (B*S)
    MergeInfo*   __restrict__ info, // (NOUT_)
    int*         __restrict__ p_out)// (NOUT_)
{
    __shared__ float gmat[W_][G_ + 4];  // g tile for this window (+pad)
    __shared__ float snorm[W_];
    __shared__ float ssim[W_ - 1];

    const int bw      = blockIdx.x;     // window id in [0, NWIN_)
    const int rowbase = bw * W_;        // flat token row base (== b*S + win*W)
    const int tid     = threadIdx.x;
    const int wave    = tid >> 5;
    const int lane    = tid & 31;
    const int mlane   = lane & 15;            // M (A) / N (B) index within tile
    const int khalf   = (lane >> 4) << 1;     // K sub-offset: 0 for lanes 0-15, 2 for 16-31
    const int gbase   = wave * 16;            // N-tile base in G dimension
    const int ncol    = gbase + mlane;        // this lane's g column

    // --- WMMA GEMM: g_tile[16x16] = x_win[16x768] * Wg[768, gbase:gbase+16] ---
    // A (16x4 f32): lane m = lane%16; VGPR0 = K+khalf, VGPR1 = K+khalf+1  (ISA 7.12.2)
    // B (4x16 f32): lane n = lane%16; VGPR0 = K+khalf, VGPR1 = K+khalf+1
    v8f acc = {};
    const float* xrow = x + (size_t)(rowbase + mlane) * D_;
    #pragma unroll 4
    for (int k0 = 0; k0 < D_; k0 += 4) {
        v2f a, b;
        a.x = xrow[k0 + khalf];
        a.y = xrow[k0 + khalf + 1];
        b.x = Wg[(size_t)(k0 + khalf) * G_ + ncol];
        b.y = Wg[(size_t)(k0 + khalf + 1) * G_ + ncol];
        // 8 args: (neg_a, A, neg_b, B, c_mod, C, reuse_a, reuse_b)
        acc = __builtin_amdgcn_wmma_f32_16x16x4_f32(
            false, a, false, b, (short)0, acc, false, false);
    }

    // C/D layout: VGPR v -> row M = v + 8*(lane/16), col N = lane%16 (ISA 7.12.2)
    const int mrow = (lane >> 4) * 8;
    #pragma unroll
    for (int v = 0; v < 8; ++v)
        gmat[mrow + v][ncol] = acc[v];
    __syncthreads();

    // --- norms per token (16 lanes) ---
    if (tid < W_) {
        float s = 0.f;
        #pragma unroll 8
        for (int k = 0; k < G_; ++k) { float t = gmat[tid][k]; s += t * t; }
        snorm[tid] = sqrtf(s);
    }
    __syncthreads();

    // --- adjacent cosine sims (15 lanes) ---
    if (tid < W_ - 1) {
        float d = 0.f;
        #pragma unroll 8
        for (int k = 0; k < G_; ++k) d += gmat[tid][k] * gmat[tid + 1][k];
        const float na = fmaxf(snorm[tid],     EPS_NORM);
        const float nb = fmaxf(snorm[tid + 1], EPS_NORM);
        ssim[tid] = d / (na * nb);
    }
    __syncthreads();

    // --- greedy non-overlapping pair selection + descriptor emission ---
    if (tid == 0) {
        // stable descending argsort of ssim (matches jnp.argsort(-sim), stable)
        int ord[W_ - 1];
        for (int i = 0; i < W_ - 1; ++i) ord[i] = i;
        for (int i = 1; i < W_ - 1; ++i) {
            const int oi = ord[i];
            const float vi = ssim[oi];
            int j = i - 1;
            while (j >= 0 && ssim[ord[j]] < vi) { ord[j + 1] = ord[j]; --j; }
            ord[j + 1] = oi;
        }
        bool used[W_];
        bool sel[W_ - 1];
        for (int i = 0; i < W_; ++i) used[i] = false;
        for (int i = 0; i < W_ - 1; ++i) sel[i] = false;
        int cnt = 0;
        for (int t = 0; t < W_ - 1; ++t) {
            const int e = ord[t];
            if (!used[e] && !used[e + 1] && cnt < R_) {
                sel[e] = true; used[e] = true; used[e + 1] = true; ++cnt;
            }
        }
        // token m removed iff sel[m-1]; kept token m merged iff sel[m]
        int out = 0;
        for (int m = 0; m < W_; ++m) {
            if (m > 0 && sel[m - 1]) continue;  // consumed by previous pair
            MergeInfo mi;
            mi.src0 = rowbase + m;
            if (m < W_ - 1 && sel[m]) {
                const float wi = snorm[m], wj = snorm[m + 1];
                const float tot = wi + wj + EPS_MERGE;
                mi.src1 = rowbase + m + 1;
                mi.w0 = wi / tot;
                mi.w1 = wj / tot;
            } else {
                mi.src1 = -1;
                mi.w0 = 1.f;
                mi.w1 = 0.f;
            }
            const int oidx = bw * WOUT_ + out;
            info[oidx]  = mi;
            p_out[oidx] = pos[rowbase + m];
            ++out;
        }
    }
}

// ---------------------------------------------------------------------------
// Kernel 2: x gather/merge. One block per output token row; 192 threads each
// move one float4 (D=768 -> 192 float4s). Pure streaming, HBM-bound.
// ---------------------------------------------------------------------------
__global__ __launch_bounds__(192) void tm_merge_x_kernel(
    const float* __restrict__ x,
    const MergeInfo* __restrict__ info,
    float* __restrict__ x_out)
{
    const int row = blockIdx.x;
    const MergeInfo mi = info[row];
    const int t = threadIdx.x;                       // 0..191
    const float4* a = (const float4*)(x + (size_t)mi.src0 * D_);
    float4 va = a[t];
    if (mi.src1 >= 0) {
        const float4* b = (const float4*)(x + (size_t)mi.src1 * D_);
        const float4 vb = b[t];
        va.x = mi.w0 * va.x + mi.w1 * vb.x;
        va.y = mi.w0 * va.y + mi.w1 * vb.y;
        va.z = mi.w0 * va.z + mi.w1 * vb.z;
        va.w = mi.w0 * va.w + mi.w1 * vb.w;
    }
    ((float4*)(x_out + (size_t)row * D_))[t] = va;
}

// ---------------------------------------------------------------------------
// Kernel 3: source gather/merge (the bandwidth hog: 268 MB read, 201 MB write).
// One block per output row of 4096 floats; 256 threads x 4 float4 each.
// ---------------------------------------------------------------------------
__global__ __launch_bounds__(256) void tm_merge_s_kernel(
    const float* __restrict__ src,
    const MergeInfo* __restrict__ info,
    float* __restrict__ s_out)
{
    const int row = blockIdx.x;
    const MergeInfo mi = info[row];
    const float4* a = (const float4*)(src + (size_t)mi.src0 * S_);
    float4* o = (float4*)(s_out + (size_t)row * S_);
    const bool mg = (mi.src1 >= 0);
    const float4* b = mg ? (const float4*)(src + (size_t)mi.src1 * S_) : a;
    #pragma unroll
    for (int t = threadIdx.x; t < S_ / 4; t += 256) {
        float4 va = a[t];
        if (mg) {
            const float4 vb = b[t];
            va.x += vb.x; va.y += vb.y; va.z += vb.z; va.w += vb.w;
        }
        o[t] = va;
    }
}

// ---------------------------------------------------------------------------
extern "C" void kernel_launch(void* const* d_in, const int* in_sizes, int n_in,
                              void* d_out, int out_size, void* d_ws, size_t ws_size,
                              hipStream_t stream) {
    const float* x      = (const float*)d_in[0];
    const float* source = (const float*)d_in[1];
    const int*   pos    = (const int*)d_in[2];
    const float* Wg     = (const float*)d_in[3];
    // d_in[4] = r (4), d_in[5] = window_size (16): baked as compile-time consts.

    float* out   = (float*)d_out;
    float* x_out = out;                                        // 12288*768
    float* s_out = out + (size_t)NOUT_ * D_;                   // 12288*4096
    int*   p_out = (int*)(out + (size_t)NOUT_ * D_ + (size_t)NOUT_ * S_);

    MergeInfo* info = (MergeInfo*)d_ws;   // 12288 * 16 B = 196 KB scratch

    tm_select_kernel<<<NWIN_, 128, 0, stream>>>(x, Wg, pos, info, p_out);
    tm_merge_x_kernel<<<NOUT_, 192, 0, stream>>>(x, info, x_out);
    tm_merge_s_kernel<<<NOUT_, 256, 0, stream>>>(source, info, s_out);
}